// DeeperGCN_86870008529302
// MI455X (gfx1250) — compile-verified
//
#include <hip/hip_runtime.h>

#define N_NODES 50000
#define N_EDGES 500000
#define HDIM 128
#define H2 256
#define NLAYERS 4
#define NGRAPHS 1000
#define MSG_EPS 1e-7f
#define LN_EPS 1e-5f

typedef __attribute__((ext_vector_type(2))) float v2f;
typedef __attribute__((ext_vector_type(8))) float v8f;

// float atomic max for non-negative values via int bit-pattern ordering
__device__ inline void atomic_max_pos(float* addr, float v) {
    atomicMax((int*)addr, __float_as_int(v));
}

// ---------------- node encoder: h = x @ Wn + bn ----------------
__global__ void k_encode_nodes(const float* __restrict__ x,
                               const float* __restrict__ W,
                               const float* __restrict__ b,
                               float* __restrict__ h) {
    int gid = blockIdx.x * blockDim.x + threadIdx.x;
    if (gid >= N_NODES * HDIM) return;
    int n = gid >> 7, c = gid & 127;
    float acc = b[c];
    const float* xr = x + (size_t)n * 9;
#pragma unroll
    for (int k = 0; k < 9; k++) acc += xr[k] * W[k * HDIM + c];
    h[gid] = acc;
}

// ---------------- pre-norm: z = relu(LN(h)*g + b), wave per node ----------------
__global__ void k_ln_relu(const float* __restrict__ h,
                          const float* __restrict__ g,
                          const float* __restrict__ b,
                          float* __restrict__ z) {
    int gid = blockIdx.x * blockDim.x + threadIdx.x;
    int n = gid >> 5, lane = gid & 31;
    if (n >= N_NODES) return;
    float v[4];
    float s = 0.f, s2 = 0.f;
#pragma unroll
    for (int j = 0; j < 4; j++) {
        v[j] = h[(size_t)n * HDIM + lane + 32 * j];
        s += v[j]; s2 += v[j] * v[j];
    }
#pragma unroll
    for (int off = 16; off > 0; off >>= 1) {
        s += __shfl_xor(s, off);
        s2 += __shfl_xor(s2, off);
    }
    float mu = s * (1.f / HDIM);
    float rstd = rsqrtf(s2 * (1.f / HDIM) - mu * mu + LN_EPS);
#pragma unroll
    for (int j = 0; j < 4; j++) {
        int c = lane + 32 * j;
        float o = (v[j] - mu) * rstd * g[c] + b[c];
        z[(size_t)n * HDIM + c] = fmaxf(o, 0.f);
    }
}

// ---------------- edge pass 1: per-(dst,ch) max of logits ----------------
__global__ void k_edge_max(const int* __restrict__ ei,
                           const float* __restrict__ eattr,
                           const float* __restrict__ We,
                           const float* __restrict__ be,
                           const float* __restrict__ zin,
                           float* __restrict__ mmax,
                           const float* __restrict__ t, int layer) {
    int gid = blockIdx.x * blockDim.x + threadIdx.x;
    int e = gid >> 5, lane = gid & 31;
    if (e >= N_EDGES) return;
    int src = ei[e], dst = ei[N_EDGES + e];
    float av = (lane < 3) ? eattr[(size_t)e * 3 + lane] : 0.f;
    float a0 = __shfl(av, 0), a1 = __shfl(av, 1), a2 = __shfl(av, 2);
    float tt = t[layer];
    const float* zs = zin + (size_t)src * HDIM;
    float* mrow = mmax + (size_t)dst * HDIM;
#pragma unroll
    for (int j = 0; j < 4; j++) {
        int c = lane + 32 * j;
        float ea = be[c] + a0 * We[c] + a1 * We[HDIM + c] + a2 * We[2 * HDIM + c];
        float msg = fmaxf(zs[c] + ea, 0.f) + MSG_EPS;
        atomic_max_pos(&mrow[c], tt * msg);
    }
}

// ---------------- edge pass 2: accumulate exp-weights and weighted msgs ----------------
__global__ void k_edge_sum(const int* __restrict__ ei,
                           const float* __restrict__ eattr,
                           const float* __restrict__ We,
                           const float* __restrict__ be,
                           const float* __restrict__ zin,
                           const float* __restrict__ mmax,
                           float* __restrict__ den,
                           float* __restrict__ aggn,
                           const float* __restrict__ t, int layer) {
    int gid = blockIdx.x * blockDim.x + threadIdx.x;
    int e = gid >> 5, lane = gid & 31;
    if (e >= N_EDGES) return;
    int src = ei[e], dst = ei[N_EDGES + e];
    float av = (lane < 3) ? eattr[(size_t)e * 3 + lane] : 0.f;
    float a0 = __shfl(av, 0), a1 = __shfl(av, 1), a2 = __shfl(av, 2);
    float tt = t[layer];
    const float* zs = zin + (size_t)src * HDIM;
    const float* mrow = mmax + (size_t)dst * HDIM;
#pragma unroll
    for (int j = 0; j < 4; j++) {
        int c = lane + 32 * j;
        float ea = be[c] + a0 * We[c] + a1 * We[HDIM + c] + a2 * We[2 * HDIM + c];
        float msg = fmaxf(zs[c] + ea, 0.f) + MSG_EPS;
        float num = __expf(tt * msg - mrow[c]);
        atomicAdd(&den[(size_t)dst * HDIM + c], num);
        atomicAdd(&aggn[(size_t)dst * HDIM + c], msg * num);
    }
}

// ---------------- finalize: xin = agg/den + zin ----------------
__global__ void k_finalize(const float* __restrict__ aggn,
                           const float* __restrict__ den,
                           const float* __restrict__ zin,
                           float* __restrict__ xin) {
    int gid = blockIdx.x * blockDim.x + threadIdx.x;
    if (gid >= N_NODES * HDIM) return;
    float d = den[gid];
    float a = (d > 0.f) ? aggn[gid] / d : 0.f;
    xin[gid] = a + zin[gid];
}

// ---------------- MLP: relu(LN(xin@W1+b1))@W2+b2, fp32 WMMA ----------------
__global__ __launch_bounds__(256)
void k_mlp_wmma(const float* __restrict__ xin,
                const float* __restrict__ W1, const float* __restrict__ b1,
                const float* __restrict__ lg, const float* __restrict__ lb,
                const float* __restrict__ W2, const float* __restrict__ b2,
                float* __restrict__ h, int add_residual) {
    __shared__ float A[16 * 132];   // padded rows (stride 132) -> no bank conflicts
    __shared__ float Z1[16 * 260];  // padded rows (stride 260)
    int tid = threadIdx.x;
    int wave = tid >> 5, lane = tid & 31;
    int hi = lane >> 4, m = lane & 15;
    int node0 = blockIdx.x * 16;

    for (int i = tid; i < 16 * HDIM; i += 256)
        A[(i >> 7) * 132 + (i & 127)] = xin[(size_t)node0 * HDIM + i];
    __syncthreads();

    // GEMM1: [16x128] @ [128x256]; wave handles col-tiles {wave, wave+8}
#pragma unroll
    for (int tt = 0; tt < 2; tt++) {
        int ncol = (wave + 8 * tt) * 16 + m;
        v8f acc = {};
        for (int k0 = 0; k0 < HDIM; k0 += 4) {
            int k = k0 + 2 * hi;
            v2f a, bm;
            a.x = A[m * 132 + k];
            a.y = A[m * 132 + k + 1];
            bm.x = W1[(size_t)k * H2 + ncol];
            bm.y = W1[(size_t)(k + 1) * H2 + ncol];
            acc = __builtin_amdgcn_wmma_f32_16x16x4_f32(false, a, false, bm,
                                                        (short)0, acc, false, false);
        }
        float bias = b1[ncol];
#pragma unroll
        for (int r = 0; r < 8; r++)
            Z1[(r + 8 * hi) * 260 + ncol] = acc[r] + bias;
    }
    __syncthreads();

    // LayerNorm(256) + relu; wave handles rows 2w, 2w+1
#pragma unroll
    for (int rr = 0; rr < 2; rr++) {
        int row = wave * 2 + rr;
        float s = 0.f, s2 = 0.f;
#pragma unroll
        for (int j = 0; j < 8; j++) {
            float v = Z1[row * 260 + lane + 32 * j];
            s += v; s2 += v * v;
        }
#pragma unroll
        for (int off = 16; off > 0; off >>= 1) {
            s += __shfl_xor(s, off);
            s2 += __shfl_xor(s2, off);
        }
        float mu = s * (1.f / H2);
        float rstd = rsqrtf(s2 * (1.f / H2) - mu * mu + LN_EPS);
#pragma unroll
        for (int j = 0; j < 8; j++) {
            int c = lane + 32 * j;
            float v = (Z1[row * 260 + c] - mu) * rstd * lg[c] + lb[c];
            Z1[row * 260 + c] = fmaxf(v, 0.f);
        }
    }
    __syncthreads();

    // GEMM2: [16x256] @ [256x128]; wave handles col-tile `wave`
    {
        int ncol = wave * 16 + m;
        v8f acc = {};
        for (int k0 = 0; k0 < H2; k0 += 4) {
            int k = k0 + 2 * hi;
            v2f a, bm;
            a.x = Z1[m * 260 + k];
            a.y = Z1[m * 260 + k + 1];
            bm.x = W2[(size_t)k * HDIM + ncol];
            bm.y = W2[(size_t)(k + 1) * HDIM + ncol];
            acc = __builtin_amdgcn_wmma_f32_16x16x4_f32(false, a, false, bm,
                                                        (short)0, acc, false, false);
        }
        float bias = b2[ncol];
#pragma unroll
        for (int r = 0; r < 8; r++) {
            size_t idx = (size_t)(node0 + r + 8 * hi) * HDIM + ncol;
            float v = acc[r] + bias;
            if (add_residual) v += h[idx];
            h[idx] = v;
        }
    }
}

// ---------------- global mean pool ----------------
__global__ void k_pool_acc(const float* __restrict__ z,
                           const int* __restrict__ batch,
                           float* __restrict__ sums,
                           float* __restrict__ counts) {
    int gid = blockIdx.x * blockDim.x + threadIdx.x;
    int n = gid >> 5, lane = gid & 31;
    if (n >= N_NODES) return;
    int g = batch[n];
#pragma unroll
    for (int j = 0; j < 4; j++) {
        int c = lane + 32 * j;
        atomicAdd(&sums[(size_t)g * HDIM + c], z[(size_t)n * HDIM + c]);
    }
    if (lane == 0) atomicAdd(&counts[g], 1.f);
}

__global__ void k_pool_div(const float* __restrict__ sums,
                           const float* __restrict__ counts,
                           float* __restrict__ out) {
    int gid = blockIdx.x * blockDim.x + threadIdx.x;
    if (gid >= NGRAPHS * HDIM) return;
    int g = gid >> 7;
    out[gid] = sums[gid] / fmaxf(counts[g], 1.f);
}

extern "C" void kernel_launch(void* const* d_in, const int* in_sizes, int n_in,
                              void* d_out, int out_size, void* d_ws, size_t ws_size,
                              hipStream_t stream) {
    const float* x     = (const float*)d_in[0];
    const int*   ei    = (const int*)d_in[1];
    const float* eattr = (const float*)d_in[2];
    const int*   batch = (const int*)d_in[3];
    const float* Wn    = (const float*)d_in[4];
    const float* bn    = (const float*)d_in[5];
    const float* We    = (const float*)d_in[6];
    const float* be    = (const float*)d_in[7];
    const float* ln_g  = (const float*)d_in[8];
    const float* ln_b  = (const float*)d_in[9];
    const float* W1    = (const float*)d_in[10];
    const float* b1    = (const float*)d_in[11];
    const float* mlg   = (const float*)d_in[12];
    const float* mlb   = (const float*)d_in[13];
    const float* W2    = (const float*)d_in[14];
    const float* b2    = (const float*)d_in[15];
    const float* t     = (const float*)d_in[16];

    size_t NH = (size_t)N_NODES * HDIM;
    float* ws     = (float*)d_ws;
    float* h      = ws;                // NH
    float* z      = h + NH;            // NH
    float* mmax   = z + NH;            // NH (aliased as xin after edge passes)
    float* den    = mmax + NH;         // NH
    float* aggn   = den + NH;          // NH
    float* sums   = aggn + NH;         // G*H
    float* counts = sums + (size_t)NGRAPHS * HDIM;  // G
    float* xin    = mmax;              // reuse: mmax dead after k_edge_sum

    dim3 b256(256);
    int gNH       = (int)((NH + 255) / 256);
    int gEdge     = (N_EDGES * 32 + 255) / 256;
    int gNodeWave = (N_NODES * 32 + 255) / 256;

    k_encode_nodes<<<gNH, b256, 0, stream>>>(x, Wn, bn, h);

    for (int layer = 0; layer < NLAYERS; layer++) {
        const float* zin;
        if (layer == 0) {
            zin = h;
        } else {
            k_ln_relu<<<gNodeWave, b256, 0, stream>>>(h, ln_g + layer * HDIM,
                                                      ln_b + layer * HDIM, z);
            zin = z;
        }
        // zero mmax, den, aggn in one contiguous memset
        hipMemsetAsync(mmax, 0, 3 * NH * sizeof(float), stream);
        k_edge_max<<<gEdge, b256, 0, stream>>>(ei, eattr, We, be, zin, mmax, t, layer);
        k_edge_sum<<<gEdge, b256, 0, stream>>>(ei, eattr, We, be, zin, mmax, den, aggn, t, layer);
        k_finalize<<<gNH, b256, 0, stream>>>(aggn, den, zin, xin);
        k_mlp_wmma<<<N_NODES / 16, b256, 0, stream>>>(
            xin,
            W1 + (size_t)layer * HDIM * H2, b1 + (size_t)layer * H2,
            mlg + (size_t)layer * H2, mlb + (size_t)layer * H2,
            W2 + (size_t)layer * H2 * HDIM, b2 + (size_t)layer * HDIM,
            h, layer > 0 ? 1 : 0);
    }

    // final norm (layer-0 params) + mean pool
    k_ln_relu<<<gNodeWave, b256, 0, stream>>>(h, ln_g, ln_b, z);
    hipMemsetAsync(sums, 0, ((size_t)NGRAPHS * HDIM + NGRAPHS) * sizeof(float), stream);
    k_pool_acc<<<gNodeWave, b256, 0, stream>>>(z, batch, sums, counts);
    k_pool_div<<<(NGRAPHS * HDIM + 255) / 256, b256, 0, stream>>>(sums, counts, (float*)d_out);
}